// GNN_70239895158819
// MI455X (gfx1250) — compile-verified
//
#include <hip/hip_runtime.h>
#include <hip/hip_bf16.h>

// ---------------------------------------------------------------------------
// GNN forward for MI455X (gfx1250, wave32, WMMA).
// Memory-bound on edge gather/scatter (~3.5 GB @ 23.3 TB/s ~ 150us floor).
// Dense GEMMs use V_WMMA_F32_16X16X4_F32 (exact fp32 on matrix pipes).
// ---------------------------------------------------------------------------

typedef __attribute__((ext_vector_type(2))) float v2f;
typedef __attribute__((ext_vector_type(8))) float v8f;

#define HDIM 128

// C[M x 128] = A[M x K] @ W[K x 128] + bias.  M % 16 == 0, K % 4 == 0.
// Block = 256 threads = 8 wave32; wave w computes columns [16w, 16w+16).
// Grid.x = M/16 (one 16-row panel per block).
__global__ __launch_bounds__(256)
void gemm_wmma_f32(const float* __restrict__ A, const float* __restrict__ W,
                   const float* __restrict__ bias, float* __restrict__ C,
                   int K, int ldc)
{
    const int lane = threadIdx.x & 31;
    const int wave = threadIdx.x >> 5;
    const int m    = lane & 15;            // A row within 16-row tile
    const int ko   = (lane >> 4) << 1;     // lanes 0-15: k+0/1, lanes 16-31: k+2/3
    const int ncol = (wave << 4) + (lane & 15);   // 0..127 output column

    const float* __restrict__ arow = A + (size_t)(blockIdx.x * 16 + m) * K;

    v8f acc = {};
    for (int kk = 0; kk < K; kk += 4) {
        v2f a, b;
        a.x = arow[kk + ko];
        a.y = arow[kk + ko + 1];
        b.x = W[(size_t)(kk + ko)     * HDIM + ncol];
        b.y = W[(size_t)(kk + ko + 1) * HDIM + ncol];
        acc = __builtin_amdgcn_wmma_f32_16x16x4_f32(
                  false, a, false, b, (short)0, acc, false, false);
    }

    const float bv    = bias[ncol];
    const int   rhalf = (lane >> 4) << 3;  // lanes 16-31 hold rows 8..15
    const size_t mg   = (size_t)blockIdx.x * 16;
#pragma unroll
    for (int r = 0; r < 8; ++r) {
        C[(mg + r + rhalf) * ldc + ncol] = acc[r] + bv;
    }
}

// One wave per edge; lane handles 4 consecutive features.
// msg = xm[src] + edge_attr @ W_edge + b_edge ; atomicAdd into agg[dst].
__global__ __launch_bounds__(256)
void edge_scatter(const float* __restrict__ xm, const float* __restrict__ ea,
                  const float* __restrict__ We, const float* __restrict__ be,
                  const int* __restrict__ src, const int* __restrict__ dst,
                  float* __restrict__ agg, int E)
{
    const int lane = threadIdx.x & 31;
    const int wave = threadIdx.x >> 5;
    const int h0   = lane << 2;

    float w[6][4], bb[4];
#pragma unroll
    for (int k = 0; k < 6; ++k)
#pragma unroll
        for (int j = 0; j < 4; ++j)
            w[k][j] = We[k * HDIM + h0 + j];
#pragma unroll
    for (int j = 0; j < 4; ++j) bb[j] = be[h0 + j];

    const int nwaves = gridDim.x << 3;
    for (int e = (blockIdx.x << 3) + wave; e < E; e += nwaves) {
        const int s = src[e];
        const int d = dst[e];
        const float a0 = ea[(size_t)e * 6 + 0];
        const float a1 = ea[(size_t)e * 6 + 1];
        const float a2 = ea[(size_t)e * 6 + 2];
        const float a3 = ea[(size_t)e * 6 + 3];
        const float a4 = ea[(size_t)e * 6 + 4];
        const float a5 = ea[(size_t)e * 6 + 5];
        const float4 xv = *(const float4*)(xm + (size_t)s * HDIM + h0);
        const float xr[4] = {xv.x, xv.y, xv.z, xv.w};
        float* __restrict__ out = agg + (size_t)d * HDIM + h0;
#pragma unroll
        for (int j = 0; j < 4; ++j) {
            float v = xr[j] + bb[j]
                    + a0 * w[0][j] + a1 * w[1][j] + a2 * w[2][j]
                    + a3 * w[3][j] + a4 * w[4][j] + a5 * w[5][j];
            unsafeAtomicAdd(out + j, v);   // global_atomic_add_f32 (L2-resident)
        }
    }
}

__global__ __launch_bounds__(256)
void relu_k(const float4* __restrict__ in, float4* __restrict__ out, size_t n4)
{
    size_t i = (size_t)blockIdx.x * blockDim.x + threadIdx.x;
    if (i < n4) {
        float4 v = in[i];
        v.x = fmaxf(v.x, 0.f); v.y = fmaxf(v.y, 0.f);
        v.z = fmaxf(v.z, 0.f); v.w = fmaxf(v.w, 0.f);
        out[i] = v;
    }
}

__global__ __launch_bounds__(256)
void relu_add_k(const float4* __restrict__ in, const float4* __restrict__ res,
                float4* __restrict__ out, size_t n4)
{
    size_t i = (size_t)blockIdx.x * blockDim.x + threadIdx.x;
    if (i < n4) {
        float4 v = in[i], r = res[i];
        v.x = fmaxf(v.x, 0.f) + r.x; v.y = fmaxf(v.y, 0.f) + r.y;
        v.z = fmaxf(v.z, 0.f) + r.z; v.w = fmaxf(v.w, 0.f) + r.w;
        out[i] = v;
    }
}

// Segment mean-pool (batch sorted): run-length accumulate locally, atomic flush
// only on graph-ID change.  Block covers 512 nodes; thread = (half, h).
__global__ __launch_bounds__(256)
void pool_k(const float* __restrict__ x, const int* __restrict__ batch,
            float* __restrict__ sums, float* __restrict__ counts, int N)
{
    const int h    = threadIdx.x & 127;
    const int half = threadIdx.x >> 7;
    int n = blockIdx.x * 512 + half;
    const int nend = min(N, blockIdx.x * 512 + 512);
    float acc = 0.f, cnt = 0.f;
    int cur = -1;
    for (; n < nend; n += 2) {
        const int g = batch[n];
        if (g != cur) {
            if (cur >= 0) {
                unsafeAtomicAdd(&sums[(size_t)cur * HDIM + h], acc);
                if (h == 0) unsafeAtomicAdd(&counts[cur], cnt);
            }
            cur = g; acc = 0.f; cnt = 0.f;
        }
        acc += x[(size_t)n * HDIM + h];
        cnt += 1.f;
    }
    if (cur >= 0) {
        unsafeAtomicAdd(&sums[(size_t)cur * HDIM + h], acc);
        if (h == 0) unsafeAtomicAdd(&counts[cur], cnt);
    }
}

// combined[g][0..127] = pooled mean; columns 128..255 already hold g-head GEMM.
__global__ __launch_bounds__(128)
void comb_pool_k(const float* __restrict__ sums, const float* __restrict__ counts,
                 float* __restrict__ comb)
{
    const int g = blockIdx.x, h = threadIdx.x;
    comb[(size_t)g * 256 + h] = sums[(size_t)g * HDIM + h] / fmaxf(counts[g], 1.f);
}

// out[g] = b_l + sum_h relu(cpre[g][h]) * W_l[h]
__global__ __launch_bounds__(128)
void head_k(const float* __restrict__ cpre, const float* __restrict__ Wl,
            const float* __restrict__ bl, float* __restrict__ out)
{
    const int g = blockIdx.x, h = threadIdx.x;
    float v = fmaxf(cpre[(size_t)g * HDIM + h], 0.f) * Wl[h];
#pragma unroll
    for (int off = 16; off; off >>= 1) v += __shfl_down(v, off, 32);
    __shared__ float red[4];
    if ((h & 31) == 0) red[h >> 5] = v;
    __syncthreads();
    if (h == 0) out[g] = red[0] + red[1] + red[2] + red[3] + bl[0];
}

extern "C" void kernel_launch(void* const* d_in, const int* in_sizes, int n_in,
                              void* d_out, int out_size, void* d_ws, size_t ws_size,
                              hipStream_t stream)
{
    const int N = 100000, E = 1000000, G = 64, F_IN = 84, GF = 10368;

    const float* x          = (const float*)d_in[0];
    const int*   eidx       = (const int*)  d_in[1];
    const float* edge_attr  = (const float*)d_in[2];
    const int*   batch      = (const int*)  d_in[3];
    const float* graph_attr = (const float*)d_in[4];
    const float* W_msg1 = (const float*)d_in[5];  const float* b_msg1 = (const float*)d_in[6];
    const float* W_edge1= (const float*)d_in[7];  const float* b_edge1= (const float*)d_in[8];
    const float* W_self1= (const float*)d_in[9];  const float* b_self1= (const float*)d_in[10];
    const float* W_msg2 = (const float*)d_in[11]; const float* b_msg2 = (const float*)d_in[12];
    const float* W_edge2= (const float*)d_in[13]; const float* b_edge2= (const float*)d_in[14];
    const float* W_msg3 = (const float*)d_in[15]; const float* b_msg3 = (const float*)d_in[16];
    const float* W_edge3= (const float*)d_in[17]; const float* b_edge3= (const float*)d_in[18];
    const float* W_g = (const float*)d_in[19]; const float* b_g = (const float*)d_in[20];
    const float* W_c = (const float*)d_in[21]; const float* b_c = (const float*)d_in[22];
    const float* W_l = (const float*)d_in[23]; const float* b_l = (const float*)d_in[24];

    const int* srcp = eidx;
    const int* dstp = eidx + E;

    // workspace carve-up
    float* ws = (float*)d_ws;
    const size_t nh = (size_t)N * HDIM;          // 12.8M floats per node buffer
    float* xm     = ws;
    float* agg    = xm  + nh;
    float* xA     = agg + nh;                    // x1, later reused as x3
    float* xB     = xA  + nh;                    // x2
    float* sums   = xB  + nh;                    // G*128
    float* counts = sums + (size_t)G * HDIM;     // G
    float* comb   = counts + G;                  // G*256
    float* cpre   = comb + (size_t)G * 256;      // G*128

    const dim3 b256(256);
    const int gemmGridN = N / 16;                // 6250 panels
    const size_t nh4 = nh / 4;
    const unsigned ewGrid = (unsigned)((nh4 + 255) / 256);
    const int edgeBlocks = 2048;                 // 16384 waves grid-striding E

    // ---- conv1: agg = x@W_self1 + b ; agg += scatter(xm[src] + ea@W_edge1) ; x1 = relu
    gemm_wmma_f32<<<gemmGridN, b256, 0, stream>>>(x, W_msg1,  b_msg1,  xm,  F_IN, HDIM);
    gemm_wmma_f32<<<gemmGridN, b256, 0, stream>>>(x, W_self1, b_self1, agg, F_IN, HDIM);
    edge_scatter<<<edgeBlocks, b256, 0, stream>>>(xm, edge_attr, W_edge1, b_edge1,
                                                  srcp, dstp, agg, E);
    relu_k<<<ewGrid, b256, 0, stream>>>((const float4*)agg, (float4*)xA, nh4);

    // ---- conv2 (self = identity): agg = x1 ; scatter ; x2 = relu(agg) + x1
    gemm_wmma_f32<<<gemmGridN, b256, 0, stream>>>(xA, W_msg2, b_msg2, xm, HDIM, HDIM);
    hipMemcpyAsync(agg, xA, nh * sizeof(float), hipMemcpyDeviceToDevice, stream);
    edge_scatter<<<edgeBlocks, b256, 0, stream>>>(xm, edge_attr, W_edge2, b_edge2,
                                                  srcp, dstp, agg, E);
    relu_add_k<<<ewGrid, b256, 0, stream>>>((const float4*)agg, (const float4*)xA,
                                            (float4*)xB, nh4);

    // ---- conv3: x3 = relu(agg) + x2   (x3 reuses xA)
    gemm_wmma_f32<<<gemmGridN, b256, 0, stream>>>(xB, W_msg3, b_msg3, xm, HDIM, HDIM);
    hipMemcpyAsync(agg, xB, nh * sizeof(float), hipMemcpyDeviceToDevice, stream);
    edge_scatter<<<edgeBlocks, b256, 0, stream>>>(xm, edge_attr, W_edge3, b_edge3,
                                                  srcp, dstp, agg, E);
    relu_add_k<<<ewGrid, b256, 0, stream>>>((const float4*)agg, (const float4*)xB,
                                            (float4*)xA, nh4);

    // ---- mean pool
    hipMemsetAsync(sums, 0, (size_t)(G * HDIM + G) * sizeof(float), stream);
    pool_k<<<(N + 511) / 512, b256, 0, stream>>>(xA, batch, sums, counts, N);

    // ---- graph head: comb[:,128:256] = graph_attr@W_g + b_g ; comb[:,0:128] = pooled
    gemm_wmma_f32<<<G / 16, b256, 0, stream>>>(graph_attr, W_g, b_g, comb + HDIM, GF, 256);
    comb_pool_k<<<G, 128, 0, stream>>>(sums, counts, comb);

    // ---- cpre = comb@W_c + b_c ; out = relu(cpre)@W_l + b_l
    gemm_wmma_f32<<<G / 16, b256, 0, stream>>>(comb, W_c, b_c, cpre, 256, HDIM);
    head_k<<<G, 128, 0, stream>>>(cpre, W_l, b_l, (float*)d_out);
}